// HGNNP_11914239279533
// MI455X (gfx1250) — compile-verified
//
#include <hip/hip_runtime.h>
#include <hip/hip_bf16.h>
#include <math.h>

#if defined(__HIP_DEVICE_COMPILE__) && !__has_builtin(__builtin_amdgcn_wmma_f32_16x16x4_f32)
#error "__builtin_amdgcn_wmma_f32_16x16x4_f32 not available for gfx1250 device pass"
#endif

typedef __attribute__((ext_vector_type(2))) float v2f;
typedef __attribute__((ext_vector_type(8))) float v8f;

#define K_A      128000
#define K_E      256000
#define K_ET     (K_A + K_E)   /* edges incl. self loops */
#define K_NMOL   4000
#define K_N      100000
#define K_M      50000
#define K_P      400000
#define K_NTEST  80000
#define K_NTRAIN 20000

enum : int { ACT_NONE = 0, ACT_RELU, ACT_SIG, ACT_BNRELU };

// ------------------------------------------------------------------
// utility kernels
// ------------------------------------------------------------------
__global__ void k_fill_f32(float* __restrict__ p, float v, size_t n) {
  size_t i = (size_t)blockIdx.x * blockDim.x + threadIdx.x;
  if (i < n) p[i] = v;
}

__global__ void k_i2f(const int* __restrict__ a, float* __restrict__ b, int n) {
  int i = blockIdx.x * blockDim.x + threadIdx.x;
  if (i < n) b[i] = (float)a[i];
}

// ------------------------------------------------------------------
// fp32 WMMA GEMM, LDS-staged B + register-preloaded A chunks.
//   - one block (8 waves) = 128 rows x 64 cols of C; all waves share ntile
//   - B chunk (64 K-rows x 64 cols) staged to LDS, swizzled so each lane's
//     two B operands are adjacent -> ds_load_2addr_b64 feeds 2 WMMAs
//   - A chunk: 16 global_load_b64 issued back-to-back into registers,
//     one wait, then 64 straight-line v_wmma per chunk
// C = act( A(MxK) @ B(KxN) + bias [+ rank-3 ybin correction] )
// ------------------------------------------------------------------
template <int ACT, bool HAS_EXT, bool DUAL>
__global__ __launch_bounds__(256) void k_gemm(
    const float* __restrict__ Am, int lda,
    const float* __restrict__ Bm, int ldb,
    const float* __restrict__ bias,
    float* __restrict__ C, float* __restrict__ C2, int ldc,
    int M, int K, int ntiles,
    const float* __restrict__ extx,   // (M,3) row-major  (y_bin)
    const float* __restrict__ extw,   // (3,ldb) row-major (tail rows of W)
    const float* __restrict__ bng, const float* __restrict__ bnb) {
  __shared__ float lds_b[64 * 64];    // 16 KB, swizzled pair layout

  const int ntile  = blockIdx.x % ntiles;
  const int mgroup = blockIdx.x / ntiles;
  const int wv     = threadIdx.x >> 5;
  const int lane   = threadIdx.x & 31;
  const int mtiles = M >> 4;
  const int mtile  = mgroup * 8 + wv;
  const bool active = (mtile < mtiles);

  const int khalf = lane >> 4;        // 0: K 0,1   1: K 2,3
  const int l15   = lane & 15;
  const int row   = (mtile << 4) + l15;
  const int rowc  = active ? row : l15;   // clamped: safe redundant loads
  const int col0g = (ntile << 6);

  const float* Ap = Am + (size_t)rowc * lda + (khalf << 1);

  // staging coords: 256 threads cover 4 rows x 64 cols per step
  const int sc  = threadIdx.x & 63;
  const int sr0 = threadIdx.x >> 6;
  // LDS store index pieces for this thread (constant across chunks)
  const int st_kh = sr0 >> 1;       // which pair within the 4-row group
  const int st_lo = sr0 & 1;        // low/high element of the pair

  v8f acc[4] = {};

  int kb = 0;
  // ---------------- full 64-row chunks, fully unrolled ----------------
  for (; kb + 64 <= K; kb += 64) {
    __syncthreads();
    {
      const float* Bp = Bm + (size_t)(kb + sr0) * ldb + col0g + sc;
#pragma unroll
      for (int k4 = 0; k4 < 16; ++k4, Bp += (size_t)ldb * 4) {
        lds_b[((((k4 << 1) + st_kh) << 6) + sc) * 2 + st_lo] = *Bp;
      }
    }
    __syncthreads();
    v2f areg[16];
#pragma unroll
    for (int k4 = 0; k4 < 16; ++k4)
      areg[k4] = *(const v2f*)(Ap + kb + (k4 << 2));
#pragma unroll
    for (int k4 = 0; k4 < 16; ++k4) {
      const float* lb = lds_b + ((((k4 << 1) + khalf) << 6) + l15) * 2;
#pragma unroll
      for (int j = 0; j < 4; ++j) {
        v2f b = *(const v2f*)(lb + (j << 5));
        acc[j] = __builtin_amdgcn_wmma_f32_16x16x4_f32(
            false, areg[k4], false, b, (short)0, acc[j], false, false);
      }
    }
  }
  // ---------------- partial tail chunk (K % 64, multiple of 4) --------
  if (kb < K) {
    const int rows = K - kb;
    __syncthreads();
    {
      const float* Bp = Bm + (size_t)(kb + sr0) * ldb + col0g + sc;
      for (int r = sr0; r < rows; r += 4, Bp += (size_t)ldb * 4) {
        int k4 = r >> 2;
        lds_b[((((k4 << 1) + st_kh) << 6) + sc) * 2 + st_lo] = *Bp;
      }
    }
    __syncthreads();
    const int nk4 = rows >> 2;
#pragma unroll 4
    for (int k4 = 0; k4 < nk4; ++k4) {
      v2f a = *(const v2f*)(Ap + kb + (k4 << 2));
      const float* lb = lds_b + ((((k4 << 1) + khalf) << 6) + l15) * 2;
#pragma unroll
      for (int j = 0; j < 4; ++j) {
        v2f b = *(const v2f*)(lb + (j << 5));
        acc[j] = __builtin_amdgcn_wmma_f32_16x16x4_f32(
            false, a, false, b, (short)0, acc[j], false, false);
      }
    }
  }

  if (!active) return;

#pragma unroll
  for (int j = 0; j < 4; ++j) {
    int   c  = col0g + (j << 4) + l15;
    float bb = bias ? bias[c] : 0.0f;
    float sc_ = 1.0f, bt = 0.0f;
    if (ACT == ACT_BNRELU) { sc_ = bng[c] * rsqrtf(1.0f + 1e-5f); bt = bnb[c]; }
    float e0 = 0.f, e1 = 0.f, e2 = 0.f;
    if (HAS_EXT) { e0 = extw[c]; e1 = extw[ldb + c]; e2 = extw[2 * ldb + c]; }
#pragma unroll
    for (int r = 0; r < 8; ++r) {
      int   rr = (mtile << 4) + r + (khalf << 3);
      float v  = acc[j][r] + bb;
      if (HAS_EXT) {
        const float* y = extx + (size_t)rr * 3;
        v += y[0] * e0 + y[1] * e1 + y[2] * e2;
      }
      if (ACT == ACT_BNRELU)      v = fmaxf(v * sc_ + bt, 0.0f);
      else if (ACT == ACT_RELU)   v = fmaxf(v, 0.0f);
      else if (ACT == ACT_SIG)    v = 1.0f / (1.0f + expf(-v));
      size_t off = (size_t)rr * ldc + c;
      C[off] = v;
      if (DUAL) C2[off] = v;
    }
  }
}

// ------------------------------------------------------------------
// GCN
// ------------------------------------------------------------------
__global__ void k_deg(const int* __restrict__ ei, float* __restrict__ deg) {
  int e = blockIdx.x * blockDim.x + threadIdx.x;
  if (e >= K_ET) return;
  int d = (e < K_E) ? ei[K_E + e] : (e - K_E);
  atomicAdd(&deg[d], 1.0f);
}

__global__ void k_rsqrt_ip(float* __restrict__ p, int n) {
  int i = blockIdx.x * blockDim.x + threadIdx.x;
  if (i < n) p[i] = rsqrtf(p[i]);
}

__global__ void k_gcn_scatter(const int* __restrict__ ei,
                              const float* __restrict__ dinv,
                              const float* __restrict__ xw,
                              float* __restrict__ h) {
  size_t idx = (size_t)blockIdx.x * blockDim.x + threadIdx.x;
  if (idx >= (size_t)K_ET * 64) return;
  int e = (int)(idx >> 6), f = (int)(idx & 63);
  int s, d;
  if (e < K_E) { s = ei[e]; d = ei[K_E + e]; } else { s = d = e - K_E; }
  atomicAdd(&h[(size_t)d * 64 + f], dinv[s] * dinv[d] * xw[(size_t)s * 64 + f]);
}

__global__ void k_bias_relu(float* __restrict__ x, const float* __restrict__ b,
                            int ld, size_t n) {
  size_t i = (size_t)blockIdx.x * blockDim.x + threadIdx.x;
  if (i < n) x[i] = fmaxf(x[i] + b[i % ld], 0.0f);
}

// ------------------------------------------------------------------
// GAT
// ------------------------------------------------------------------
__device__ __forceinline__ unsigned f2key(float f) {
  unsigned u = __float_as_uint(f);
  return (u & 0x80000000u) ? ~u : (u | 0x80000000u);
}
__device__ __forceinline__ float key2f(unsigned k) {
  return (k & 0x80000000u) ? __uint_as_float(k ^ 0x80000000u)
                           : __uint_as_float(~k);
}
__device__ __forceinline__ float leaky(float x) { return x > 0.f ? x : 0.2f * x; }

__global__ void k_gat_al(const float* __restrict__ xh,
                         const float* __restrict__ asrc,
                         const float* __restrict__ adst,
                         float* __restrict__ als, float* __restrict__ ald) {
  int i = blockIdx.x * blockDim.x + threadIdx.x;
  if (i >= K_A * 2) return;
  int atom = i >> 1, hd = i & 1;
  const float* x = xh + (size_t)atom * 256 + hd * 128;
  const float* ws_ = asrc + hd * 128;
  const float* wd_ = adst + hd * 128;
  float ss = 0.f, sd = 0.f;
  for (int c = 0; c < 128; ++c) { float v = x[c]; ss += v * ws_[c]; sd += v * wd_[c]; }
  als[i] = ss; ald[i] = sd;
}

__global__ void k_gat_max(const int* __restrict__ ei,
                          const float* __restrict__ als,
                          const float* __restrict__ ald,
                          unsigned* __restrict__ mx) {
  int i = blockIdx.x * blockDim.x + threadIdx.x;
  if (i >= K_ET * 2) return;
  int e = i >> 1, hd = i & 1, s, d;
  if (e < K_E) { s = ei[e]; d = ei[K_E + e]; } else { s = d = e - K_E; }
  float v = leaky(als[s * 2 + hd] + ald[d * 2 + hd]);
  atomicMax(&mx[d * 2 + hd], f2key(v));
}

__global__ void k_gat_expsum(const int* __restrict__ ei,
                             const float* __restrict__ als,
                             const float* __restrict__ ald,
                             const unsigned* __restrict__ mx,
                             float* __restrict__ pbuf, float* __restrict__ ssum) {
  int i = blockIdx.x * blockDim.x + threadIdx.x;
  if (i >= K_ET * 2) return;
  int e = i >> 1, hd = i & 1, s, d;
  if (e < K_E) { s = ei[e]; d = ei[K_E + e]; } else { s = d = e - K_E; }
  float v = leaky(als[s * 2 + hd] + ald[d * 2 + hd]);
  float p = expf(v - key2f(mx[d * 2 + hd]));
  pbuf[i] = p;
  atomicAdd(&ssum[d * 2 + hd], p);
}

__global__ void k_gat_scatter(const int* __restrict__ ei,
                              const float* __restrict__ pbuf,
                              const float* __restrict__ ssum,
                              const float* __restrict__ xh,
                              float* __restrict__ g) {
  size_t idx = (size_t)blockIdx.x * blockDim.x + threadIdx.x;
  if (idx >= (size_t)K_ET * 256) return;
  int e = (int)(idx >> 8), f = (int)(idx & 255), hd = f >> 7;
  int s, d;
  if (e < K_E) { s = ei[e]; d = ei[K_E + e]; } else { s = d = e - K_E; }
  float alpha = pbuf[e * 2 + hd] / ssum[d * 2 + hd];
  atomicAdd(&g[(size_t)d * 256 + f], alpha * xh[(size_t)s * 256 + f]);
}

// ------------------------------------------------------------------
// mol pooling
// ------------------------------------------------------------------
__global__ void k_cnt(const int* __restrict__ mol, float* __restrict__ cnt) {
  int a = blockIdx.x * blockDim.x + threadIdx.x;
  if (a < K_A) atomicAdd(&cnt[mol[a]], 1.0f);
}

__global__ void k_mol_scatter(const int* __restrict__ mol,
                              const float* __restrict__ g,
                              float* __restrict__ mat) {
  size_t idx = (size_t)blockIdx.x * blockDim.x + threadIdx.x;
  if (idx >= (size_t)K_A * 256) return;
  int a = (int)(idx >> 8), f = (int)(idx & 255);
  atomicAdd(&mat[(size_t)mol[a] * 256 + f], g[idx]);
}

__global__ void k_mol_final(float* __restrict__ mat, const float* __restrict__ cnt,
                            const float* __restrict__ gat_b) {
  size_t i = (size_t)blockIdx.x * blockDim.x + threadIdx.x;
  if (i >= (size_t)K_NMOL * 256) return;
  int m = (int)(i >> 8), f = (int)(i & 255);
  mat[i] = mat[i] / cnt[m] + gat_b[f];
}

// ------------------------------------------------------------------
// v2v mean
// ------------------------------------------------------------------
__global__ void k_pair_deg(const int* __restrict__ pv, const int* __restrict__ pe,
                           float* __restrict__ dv, float* __restrict__ de) {
  int p = blockIdx.x * blockDim.x + threadIdx.x;
  if (p >= K_P) return;
  atomicAdd(&dv[pv[p]], 1.0f);
  atomicAdd(&de[pe[p]], 1.0f);
}

__global__ void k_recip_max1(float* __restrict__ x, int n) {
  int i = blockIdx.x * blockDim.x + threadIdx.x;
  if (i < n) x[i] = 1.0f / fmaxf(x[i], 1.0f);
}

__global__ void k_pair_fwd(const int* __restrict__ pv, const int* __restrict__ pe,
                           const float* __restrict__ X, float* __restrict__ ef) {
  size_t idx = (size_t)blockIdx.x * blockDim.x + threadIdx.x;
  if (idx >= (size_t)K_P * 256) return;
  int p = (int)(idx >> 8), f = (int)(idx & 255);
  atomicAdd(&ef[(size_t)pe[p] * 256 + f], X[(size_t)pv[p] * 256 + f]);
}

__global__ void k_scale_rows(float* __restrict__ x, const float* __restrict__ r, size_t n) {
  size_t i = (size_t)blockIdx.x * blockDim.x + threadIdx.x;
  if (i < n) x[i] *= r[i >> 8];
}

__global__ void k_pair_bwd(const int* __restrict__ pv, const int* __restrict__ pe,
                           const float* __restrict__ ef, float* __restrict__ X) {
  size_t idx = (size_t)blockIdx.x * blockDim.x + threadIdx.x;
  if (idx >= (size_t)K_P * 256) return;
  int p = (int)(idx >> 8), f = (int)(idx & 255);
  atomicAdd(&X[(size_t)pv[p] * 256 + f], ef[(size_t)pe[p] * 256 + f]);
}

__global__ void k_v2v_relu(float* __restrict__ X, const float* __restrict__ rdv, size_t n) {
  size_t i = (size_t)blockIdx.x * blockDim.x + threadIdx.x;
  if (i < n) X[i] = fmaxf(X[i] * rdv[i >> 8], 0.0f);
}

__global__ void k_feat_final(float* __restrict__ F, const float* __restrict__ rdv,
                             const float* __restrict__ mat, const int* __restrict__ n2m) {
  size_t i = (size_t)blockIdx.x * blockDim.x + threadIdx.x;
  if (i >= (size_t)K_N * 256) return;
  int node = (int)(i >> 8), f = (int)(i & 255);
  F[i] = F[i] * rdv[node] + mat[(size_t)n2m[node] * 256 + f];
}

// ------------------------------------------------------------------
// logits (64 -> 3)
// ------------------------------------------------------------------
__global__ void k_logits(const float* __restrict__ h3, const float* __restrict__ w,
                         const float* __restrict__ b, float* __restrict__ out) {
  int r = blockIdx.x * blockDim.x + threadIdx.x;
  if (r >= K_NTEST) return;
  const float* hr = h3 + (size_t)r * 64;
  float a0 = b[0], a1 = b[1], a2 = b[2];
  for (int k = 0; k < 64; ++k) {
    float v = hr[k];
    a0 += v * w[k * 3 + 0];
    a1 += v * w[k * 3 + 1];
    a2 += v * w[k * 3 + 2];
  }
  out[r * 3 + 0] = a0; out[r * 3 + 1] = a1; out[r * 3 + 2] = a2;
}

// ------------------------------------------------------------------
// host side
// ------------------------------------------------------------------
static inline unsigned blks(size_t n) { return (unsigned)((n + 255) / 256); }
static inline unsigned gemm_grid(int M, int N) {
  int mt = M / 16; int mg = (mt + 7) / 8; return (unsigned)(mg * (N / 64));
}

extern "C" void kernel_launch(void* const* d_in, const int* in_sizes, int n_in,
                              void* d_out, int out_size, void* d_ws, size_t ws_size,
                              hipStream_t stream) {
  (void)in_sizes; (void)n_in; (void)out_size; (void)ws_size;
  const float* atom_x  = (const float*)d_in[0];
  const float* gcn_w   = (const float*)d_in[1];
  const float* gcn_b   = (const float*)d_in[2];
  const float* gat_w   = (const float*)d_in[3];
  const float* gat_as  = (const float*)d_in[4];
  const float* gat_ad  = (const float*)d_in[5];
  const float* gat_b   = (const float*)d_in[6];
  const float* feature = (const float*)d_in[7];
  const float* th1_w   = (const float*)d_in[8];
  const float* th1_b   = (const float*)d_in[9];
  const float* th2_w   = (const float*)d_in[10];
  const float* th2_b   = (const float*)d_in[11];
  const float* fcmu_w  = (const float*)d_in[12];
  const float* fcmu_b  = (const float*)d_in[13];
  const float* fclv_w  = (const float*)d_in[14];
  const float* fclv_b  = (const float*)d_in[15];
  const float* lin3_w  = (const float*)d_in[16];
  const float* lin3_b  = (const float*)d_in[17];
  const float* bn3_g   = (const float*)d_in[18];
  const float* bn3_b   = (const float*)d_in[19];
  const float* cls_w   = (const float*)d_in[20];
  const float* cls_b   = (const float*)d_in[21];
  const float* dec_w1  = (const float*)d_in[22];
  const float* dec_b1  = (const float*)d_in[23];
  const float* dec_w2  = (const float*)d_in[24];
  const float* dec_b2  = (const float*)d_in[25];
  const float* y_bin   = (const float*)d_in[26];
  const int*   ei      = (const int*)d_in[27];
  const int*   mol_ids = (const int*)d_in[28];
  const int*   n2m     = (const int*)d_in[29];
  const int*   pv      = (const int*)d_in[30];
  const int*   pe      = (const int*)d_in[31];
  const int*   y_tgt   = (const int*)d_in[32];

  float* out = (float*)d_out;
  float* ws  = (float*)d_ws;

  const size_t OFF_MU  = 0;
  const size_t OFF_LV  = OFF_MU + (size_t)K_NTEST * 64;
  const size_t OFF_Z   = OFF_LV + (size_t)K_NTEST * 64;
  const size_t OFF_LG  = OFF_Z + (size_t)K_NTEST * 64;
  const size_t OFF_XR  = OFF_LG + (size_t)K_NTEST * 3;
  const size_t OFF_Y   = OFF_XR + (size_t)K_NTEST * 512;
  const size_t OFF_FT  = OFF_Y + (size_t)K_NTEST;

  size_t o = 0;
  auto alloc = [&](size_t n) { size_t r = o; o += (n + 63) & ~(size_t)63; return r; };
  const size_t W_DEG = alloc(K_A);
  const size_t W_ALS = alloc((size_t)K_A * 2);
  const size_t W_ALD = alloc((size_t)K_A * 2);
  const size_t W_MX  = alloc((size_t)K_A * 2);
  const size_t W_SS  = alloc((size_t)K_A * 2);
  const size_t W_PB  = alloc((size_t)K_ET * 2);
  const size_t W_CNT = alloc(K_NMOL);
  const size_t W_RDE = alloc(K_M);
  const size_t W_RDV = alloc(K_N);
  const size_t W_MAT = alloc((size_t)K_NMOL * 256);
  const size_t W_XW  = alloc((size_t)K_A * 64);
  const size_t W_H   = alloc((size_t)K_A * 64);
  const size_t U_A   = alloc((size_t)K_A * 256);
  const size_t U_B   = alloc((size_t)K_A * 256);
  const size_t W_EF  = W_XW;
  const size_t W_H3  = W_XW;
  const size_t W_HD  = U_A;
  (void)W_H;

  k_gemm<ACT_NONE, false, false><<<gemm_grid(K_A, 64), 256, 0, stream>>>(
      atom_x, 48, gcn_w, 64, nullptr, ws + W_XW, nullptr, 64, K_A, 48, 1,
      nullptr, nullptr, nullptr, nullptr);
  k_fill_f32<<<blks(K_A), 256, 0, stream>>>(ws + W_DEG, 0.f, K_A);
  k_deg<<<blks(K_ET), 256, 0, stream>>>(ei, ws + W_DEG);
  k_rsqrt_ip<<<blks(K_A), 256, 0, stream>>>(ws + W_DEG, K_A);
  k_fill_f32<<<blks((size_t)K_A * 64), 256, 0, stream>>>(ws + W_H, 0.f, (size_t)K_A * 64);
  k_gcn_scatter<<<blks((size_t)K_ET * 64), 256, 0, stream>>>(ei, ws + W_DEG, ws + W_XW, ws + W_H);
  k_bias_relu<<<blks((size_t)K_A * 64), 256, 0, stream>>>(ws + W_H, gcn_b, 64, (size_t)K_A * 64);

  k_gemm<ACT_NONE, false, false><<<gemm_grid(K_A, 256), 256, 0, stream>>>(
      ws + W_H, 64, gat_w, 256, nullptr, ws + U_A, nullptr, 256, K_A, 64, 4,
      nullptr, nullptr, nullptr, nullptr);
  k_gat_al<<<blks((size_t)K_A * 2), 256, 0, stream>>>(ws + U_A, gat_as, gat_ad, ws + W_ALS, ws + W_ALD);
  k_fill_f32<<<blks((size_t)K_A * 2), 256, 0, stream>>>(ws + W_MX, 0.f, (size_t)K_A * 2);
  k_fill_f32<<<blks((size_t)K_A * 2), 256, 0, stream>>>(ws + W_SS, 0.f, (size_t)K_A * 2);
  k_gat_max<<<blks((size_t)K_ET * 2), 256, 0, stream>>>(ei, ws + W_ALS, ws + W_ALD, (unsigned*)(ws + W_MX));
  k_gat_expsum<<<blks((size_t)K_ET * 2), 256, 0, stream>>>(ei, ws + W_ALS, ws + W_ALD,
                                                           (const unsigned*)(ws + W_MX),
                                                           ws + W_PB, ws + W_SS);
  k_fill_f32<<<blks((size_t)K_A * 256), 256, 0, stream>>>(ws + U_B, 0.f, (size_t)K_A * 256);
  k_gat_scatter<<<blks((size_t)K_ET * 256), 256, 0, stream>>>(ei, ws + W_PB, ws + W_SS, ws + U_A, ws + U_B);

  k_fill_f32<<<blks(K_NMOL), 256, 0, stream>>>(ws + W_CNT, 0.f, K_NMOL);
  k_fill_f32<<<blks((size_t)K_NMOL * 256), 256, 0, stream>>>(ws + W_MAT, 0.f, (size_t)K_NMOL * 256);
  k_cnt<<<blks(K_A), 256, 0, stream>>>(mol_ids, ws + W_CNT);
  k_mol_scatter<<<blks((size_t)K_A * 256), 256, 0, stream>>>(mol_ids, ws + U_B, ws + W_MAT);
  k_mol_final<<<blks((size_t)K_NMOL * 256), 256, 0, stream>>>(ws + W_MAT, ws + W_CNT, gat_b);

  k_gemm<ACT_RELU, false, false><<<gemm_grid(K_N, 256), 256, 0, stream>>>(
      feature, 512, th1_w, 256, th1_b, ws + U_A, nullptr, 256, K_N, 512, 4,
      nullptr, nullptr, nullptr, nullptr);
  k_fill_f32<<<blks(K_M), 256, 0, stream>>>(ws + W_RDE, 0.f, K_M);
  k_fill_f32<<<blks(K_N), 256, 0, stream>>>(ws + W_RDV, 0.f, K_N);
  k_pair_deg<<<blks(K_P), 256, 0, stream>>>(pv, pe, ws + W_RDV, ws + W_RDE);
  k_recip_max1<<<blks(K_M), 256, 0, stream>>>(ws + W_RDE, K_M);
  k_recip_max1<<<blks(K_N), 256, 0, stream>>>(ws + W_RDV, K_N);

  k_fill_f32<<<blks((size_t)K_M * 256), 256, 0, stream>>>(ws + W_EF, 0.f, (size_t)K_M * 256);
  k_pair_fwd<<<blks((size_t)K_P * 256), 256, 0, stream>>>(pv, pe, ws + U_A, ws + W_EF);
  k_scale_rows<<<blks((size_t)K_M * 256), 256, 0, stream>>>(ws + W_EF, ws + W_RDE, (size_t)K_M * 256);
  k_fill_f32<<<blks((size_t)K_N * 256), 256, 0, stream>>>(ws + U_B, 0.f, (size_t)K_N * 256);
  k_pair_bwd<<<blks((size_t)K_P * 256), 256, 0, stream>>>(pv, pe, ws + W_EF, ws + U_B);
  k_v2v_relu<<<blks((size_t)K_N * 256), 256, 0, stream>>>(ws + U_B, ws + W_RDV, (size_t)K_N * 256);

  k_gemm<ACT_RELU, false, false><<<gemm_grid(K_N, 256), 256, 0, stream>>>(
      ws + U_B, 256, th2_w, 256, th2_b, ws + U_A, nullptr, 256, K_N, 256, 4,
      nullptr, nullptr, nullptr, nullptr);

  k_fill_f32<<<blks((size_t)K_M * 256), 256, 0, stream>>>(ws + W_EF, 0.f, (size_t)K_M * 256);
  k_pair_fwd<<<blks((size_t)K_P * 256), 256, 0, stream>>>(pv, pe, ws + U_A, ws + W_EF);
  k_scale_rows<<<blks((size_t)K_M * 256), 256, 0, stream>>>(ws + W_EF, ws + W_RDE, (size_t)K_M * 256);
  k_fill_f32<<<blks((size_t)K_N * 256), 256, 0, stream>>>(out + OFF_FT, 0.f, (size_t)K_N * 256);
  k_pair_bwd<<<blks((size_t)K_P * 256), 256, 0, stream>>>(pv, pe, ws + W_EF, out + OFF_FT);
  k_feat_final<<<blks((size_t)K_N * 256), 256, 0, stream>>>(out + OFF_FT, ws + W_RDV, ws + W_MAT, n2m);

  const float* ft = out + OFF_FT + (size_t)K_NTRAIN * 256;
  k_gemm<ACT_NONE, true, true><<<gemm_grid(K_NTEST, 64), 256, 0, stream>>>(
      ft, 256, fcmu_w, 64, fcmu_b, out + OFF_MU, out + OFF_Z, 64, K_NTEST, 256, 1,
      y_bin, fcmu_w + (size_t)256 * 64, nullptr, nullptr);
  k_gemm<ACT_NONE, true, false><<<gemm_grid(K_NTEST, 64), 256, 0, stream>>>(
      ft, 256, fclv_w, 64, fclv_b, out + OFF_LV, nullptr, 64, K_NTEST, 256, 1,
      y_bin, fclv_w + (size_t)256 * 64, nullptr, nullptr);
  k_gemm<ACT_BNRELU, false, false><<<gemm_grid(K_NTEST, 64), 256, 0, stream>>>(
      out + OFF_Z, 64, lin3_w, 64, lin3_b, ws + W_H3, nullptr, 64, K_NTEST, 64, 1,
      nullptr, nullptr, bn3_g, bn3_b);
  k_logits<<<blks(K_NTEST), 256, 0, stream>>>(ws + W_H3, cls_w, cls_b, out + OFF_LG);

  k_gemm<ACT_RELU, true, false><<<gemm_grid(K_NTEST, 512), 256, 0, stream>>>(
      out + OFF_Z, 64, dec_w1, 512, dec_b1, ws + W_HD, nullptr, 512, K_NTEST, 64, 8,
      y_bin, dec_w1 + (size_t)64 * 512, nullptr, nullptr);
  k_gemm<ACT_SIG, false, false><<<gemm_grid(K_NTEST, 512), 256, 0, stream>>>(
      ws + W_HD, 512, dec_w2, 512, dec_b2, out + OFF_XR, nullptr, 512, K_NTEST, 512, 8,
      nullptr, nullptr, nullptr, nullptr);
  k_i2f<<<blks(K_NTEST), 256, 0, stream>>>(y_tgt, out + OFF_Y, K_NTEST);
}